// RotationModule_62654982914426
// MI455X (gfx1250) — compile-verified
//
#include <hip/hip_runtime.h>
#include <math.h>

#define B_   2
#define L_   2048
#define NE_  768
#define NH_  12
#define NB_  16
#define BS_  4
#define K_   8
#define DT_  4
#define ROWS (B_ * L_)      // 4096
#define JDIM (NH_ * NB_)    // 192
#define NELEM (ROWS * JDIM) // 786432

typedef __attribute__((ext_vector_type(2))) float v2f;
typedef __attribute__((ext_vector_type(8))) float v8f;

// workspace layout (float indices)
#define WS_SCALE 0      // int at float slot 0
#define WS_BN    8      // 16 floats: block Frobenius norms
#define WS_MATP  256    // 2048 floats: matp[(k*16+n)*16 + e] = S_skew^k/k!
#define WS_THETA 4096   // 786432 floats: theta (B,L,NH,NB) flat

// ---------------- kernel 1: tiny precompute (1 block) ----------------
__global__ __launch_bounds__(256) void k_prep(const float* __restrict__ S,
                                              float* __restrict__ ws) {
  __shared__ float sskew[256];
  __shared__ float acc[256];
  __shared__ float red[256];
  int t = threadIdx.x;
  int n = t >> 4, e = t & 15, i = e >> 2, j = e & 3;

  float sv  = S[n * 16 + i * 4 + j];
  float svT = S[n * 16 + j * 4 + i];
  sskew[t] = sv - svT;                       // S - S^T
  float a0 = (i == j) ? 1.0f : 0.0f;         // identity = S^0
  acc[t] = a0;
  ws[WS_MATP + (0 * 16 + n) * 16 + e] = a0;  // /0! = 1
  __syncthreads();

  float fact = 1.0f;
  for (int k = 1; k < K_; ++k) {
    float v = 0.0f;
    #pragma unroll
    for (int m = 0; m < 4; ++m)
      v += sskew[n * 16 + i * 4 + m] * acc[n * 16 + m * 4 + j];
    __syncthreads();
    acc[t] = v;
    __syncthreads();
    fact *= (float)k;
    ws[WS_MATP + (k * 16 + n) * 16 + e] = v / fact;
  }

  // block norms from clipped-S skew
  const float TWO_PI = 6.283185307179586f;
  float c1 = fminf(fmaxf(sv, 0.0f), TWO_PI);
  float c2 = fminf(fmaxf(svT, 0.0f), TWO_PI);
  float d = c1 - c2;
  red[t] = d * d;
  __syncthreads();
  if (e == 0) {
    float s = 0.0f;
    #pragma unroll
    for (int q = 0; q < 16; ++q) s += red[n * 16 + q];
    ws[WS_BN + n] = sqrtf(s);
  }
  if (t == 0) ((int*)ws)[WS_SCALE] = 1;  // s >= 1 always
}

// ------- kernel 2: theta = tanh(alpha*((x@W1+b1)@W2)), scale max -------
__global__ __launch_bounds__(128) void k_theta(const float* __restrict__ x,
                                               const float* __restrict__ W1,
                                               const float* __restrict__ b1,
                                               const float* __restrict__ W2,
                                               const float* __restrict__ alpha,
                                               float* __restrict__ ws) {
  __shared__ float shW1[NE_ * DT_ + 16];  // pad so unconditional reads stay in-bounds
  __shared__ float shW2[DT_ * JDIM];
  __shared__ float shT1[4][16][DT_];
  __shared__ float shBN[NB_];
  __shared__ float shAl[NB_];
  __shared__ float shMax[4];

  int t = threadIdx.x;
  for (int q = t; q < NE_ * DT_ + 16; q += 128)
    shW1[q] = (q < NE_ * DT_) ? W1[q] : 0.0f;
  for (int q = t; q < DT_ * JDIM; q += 128) shW2[q] = W2[q];
  if (t < NB_) { shBN[t] = ws[WS_BN + t]; shAl[t] = alpha[t]; }
  __syncthreads();

  int wave = t >> 5, lane = t & 31;
  int tile = blockIdx.x * 4 + wave;
  int m0 = tile * 16;
  int half = lane >> 4;     // which K-pair half of the wave
  int lrow = lane & 15;     // A-row (lanes 0-15 and 16-31 both cover M=0..15)
  int kofs = half * 2;

  // --- x(16x768) @ W1(768x4, padded to N=16) via v_wmma_f32_16x16x4_f32 ---
  v8f acc = {};
  const float* xrow = x + (size_t)(m0 + lrow) * NE_;
  bool bvalid = (lrow < DT_);
  for (int kb = 0; kb < NE_; kb += 4) {
    v2f a;
    a.x = xrow[kb + kofs];
    a.y = xrow[kb + kofs + 1];
    v2f bm;
    float f0 = shW1[(kb + kofs) * DT_ + lrow];      // pad covers lrow>=4
    float f1 = shW1[(kb + kofs + 1) * DT_ + lrow];
    bm.x = bvalid ? f0 : 0.0f;
    bm.y = bvalid ? f1 : 0.0f;
    acc = __builtin_amdgcn_wmma_f32_16x16x4_f32(false, a, false, bm,
                                                (short)0, acc, false, false);
  }
  // C layout: lane holds N=lane%16 for rows M = v + 8*(lane/16), v=0..7
  if (bvalid) {
    float bb = b1[lrow];
    #pragma unroll
    for (int v = 0; v < 8; ++v)
      shT1[wave][half * 8 + v][lrow] = acc[v] + bb;
  }
  __syncthreads();

  // --- second tiny GEMM (x@W1+b1)@W2, tanh, store theta, scale ---
  float smax = 1.0f;
  for (int idx = lane; idx < 16 * JDIM; idx += 32) {
    int r = idx / JDIM, jj = idx - r * JDIM;
    float th = 0.0f;
    #pragma unroll
    for (int d = 0; d < DT_; ++d)
      th = fmaf(shT1[wave][r][d], shW2[d * JDIM + jj], th);
    int n = jj & 15;
    th = tanhf(shAl[n] * th);
    ws[WS_THETA + (size_t)(m0 + r) * JDIM + jj] = th;
    float nrm = fabsf(th) * shBN[n];
    float s = ceilf(log2f(nrm));   // log2(0) = -inf -> handled by select below
    s = (s > 0.0f) ? s : 1.0f;
    smax = fmaxf(smax, s);
  }
  #pragma unroll
  for (int o = 16; o > 0; o >>= 1)
    smax = fmaxf(smax, __shfl_xor(smax, o, 32));
  if (lane == 0) shMax[wave] = smax;
  __syncthreads();
  if (t == 0) {
    float m = fmaxf(fmaxf(shMax[0], shMax[1]), fmaxf(shMax[2], shMax[3]));
    atomicMax((int*)ws + WS_SCALE, (int)m);
  }
}

// ------- kernel 3: Taylor bind + scaling-and-squaring (store-bound) -------
__global__ __launch_bounds__(256) void k_expm(const float* __restrict__ ws,
                                              float* __restrict__ out) {
  __shared__ float shP[K_ * NB_ * 17];  // padded rows: (k*16+n)*17 + e
  int t = threadIdx.x;
  for (int q = t; q < K_ * NB_ * 16; q += 256) {
    int kn = q >> 4, e = q & 15;
    shP[kn * 17 + e] = ws[WS_MATP + q];
  }
  __syncthreads();

  int ms = ((const int*)ws)[WS_SCALE];
  float inv = exp2f(-(float)ms);        // exact power-of-two scale
  size_t g = (size_t)blockIdx.x * 256 + t;

  float th = ws[WS_THETA + g] * inv;
  int n = (int)(g & 15);
  float a = fabsf(th) + 1e-6f;          // matches reference's log(|th|+1e-6)
  bool neg = (th < 0.0f);

  float m[16];
  #pragma unroll
  for (int e = 0; e < 16; ++e) m[e] = 0.0f;
  float p = 1.0f;
  #pragma unroll
  for (int k = 0; k < K_; ++k) {
    float coef = (neg && (k & 1)) ? -p : p;
    const float* row = &shP[(k * 16 + n) * 17];
    #pragma unroll
    for (int e = 0; e < 16; ++e) m[e] = fmaf(coef, row[e], m[e]);
    p *= a;
  }

  int nsq = (ms < 4) ? ms : 4;          // MAX_SQ static bound
  for (int i = 0; i < nsq; ++i) {
    float c[16];
    #pragma unroll
    for (int r = 0; r < 4; ++r)
      #pragma unroll
      for (int cc = 0; cc < 4; ++cc) {
        float v = 0.0f;
        #pragma unroll
        for (int q2 = 0; q2 < 4; ++q2)
          v = fmaf(m[r * 4 + q2], m[q2 * 4 + cc], v);
        c[r * 4 + cc] = v;
      }
    #pragma unroll
    for (int e = 0; e < 16; ++e) m[e] = c[e];
  }

  float4* o = (float4*)(out + g * 16);
  o[0] = make_float4(m[0],  m[1],  m[2],  m[3]);
  o[1] = make_float4(m[4],  m[5],  m[6],  m[7]);
  o[2] = make_float4(m[8],  m[9],  m[10], m[11]);
  o[3] = make_float4(m[12], m[13], m[14], m[15]);
}

extern "C" void kernel_launch(void* const* d_in, const int* in_sizes, int n_in,
                              void* d_out, int out_size, void* d_ws, size_t ws_size,
                              hipStream_t stream) {
  const float* x  = (const float*)d_in[0];
  const float* W1 = (const float*)d_in[1];
  const float* b1 = (const float*)d_in[2];
  const float* W2 = (const float*)d_in[3];
  const float* al = (const float*)d_in[4];
  const float* S  = (const float*)d_in[5];
  float* ws  = (float*)d_ws;
  float* out = (float*)d_out;

  hipLaunchKernelGGL(k_prep,  dim3(1),            dim3(256), 0, stream, S, ws);
  hipLaunchKernelGGL(k_theta, dim3(ROWS / 64),    dim3(128), 0, stream,
                     x, W1, b1, W2, al, ws);
  hipLaunchKernelGGL(k_expm,  dim3(NELEM / 256),  dim3(256), 0, stream, ws, out);
}